// NN_GAM_13254269075985
// MI455X (gfx1250) — compile-verified
//
#include <hip/hip_runtime.h>

typedef __attribute__((ext_vector_type(2))) float v2f;
typedef __attribute__((ext_vector_type(8))) float v8f;

#define NFEAT   64
#define HID     16
#define NSAMP   65536
#define WDIM    1024          /* H1*D = row stride of W2/W3 */
#define ROWPAD  68            /* padded LDS row stride (floats): bank (4*s+d)%64 */
#define FPW     4             /* features per wave (16 waves x 4 = 64) */
#define TPW     8             /* 16-sample tiles per workgroup */
#define NBLK    (NSAMP / (16 * TPW))   /* 512 blocks */

__device__ __forceinline__ float eluf(float z) {
    // elu(z) = max(z,0) + exp(min(z,0)) - 1   (branch-free)
    float t = fminf(z, 0.0f);
    float e = __expf(t) - 1.0f;
    return fmaxf(z, 0.0f) + e;
}

// swap lane halves (lanes 0-15 <-> 16-31): ds_swizzle group-of-32, xor=0x10, and=0x1f
__device__ __forceinline__ float swap16(float x) {
    return __int_as_float(__builtin_amdgcn_ds_swizzle(__float_as_int(x), 0x401F));
}

__global__ __launch_bounds__(512, 1) void gam_wmma_kernel(
    const float* __restrict__ x,    // (N, 64)
    const float* __restrict__ W1,   // (64, 1024)
    const float* __restrict__ W2,   // (1024, 1024)
    const float* __restrict__ W3,   // (1024, 1024)
    const float* __restrict__ W4,   // (1024, 64)
    const float* __restrict__ Wp,   // (64, 1)
    const float* __restrict__ bp,   // (1,)
    float* __restrict__ pred_out,   // (N,)
    float* __restrict__ gam_out)    // (N, 64)
{
    __shared__ float xs[16 * ROWPAD];
    __shared__ float os[16 * ROWPAD];
    __shared__ float preds[16];

    const int tid  = threadIdx.x;
    const int lane = tid & 31;
    const int wv   = tid >> 5;      // wave id 0..15
    const int hl   = lane >> 4;     // half of wave (0/1)
    const int ln   = lane & 15;     // sample/N index within fragment

    // ---- per-feature weights, resident in registers for the whole kernel ----
    // A-fragment layout (f32 16x16x4, frag k): reg v, lane l -> (M = l%16, K = 4k+v+2*(l/16))
    v2f   w2a[FPW][4], w3a[FPW][4];
    float w1r[FPW][8];              // B-frag order: r=2k+v holds hidden h = 4k+v+2*hl
    float w4r[FPW][8];              // C-layout order: reg v holds hidden h = v+8*hl
    float wpr[FPW];
    #pragma unroll
    for (int f = 0; f < FPW; ++f) {
        const int d = wv * FPW + f;
        #pragma unroll
        for (int k = 0; k < 4; ++k) {
            #pragma unroll
            for (int v = 0; v < 2; ++v) {
                const int h = 4 * k + v + 2 * hl;
                w1r[f][2 * k + v] = W1[d * WDIM + d * HID + h];
                // A = W^T : A[M=ln (out n), K=h (in)] = W[d*16+h, d*16+ln]
                w2a[f][k][v] = W2[(d * HID + h) * WDIM + d * HID + ln];
                w3a[f][k][v] = W3[(d * HID + h) * WDIM + d * HID + ln];
            }
        }
        #pragma unroll
        for (int v = 0; v < 8; ++v)
            w4r[f][v] = W4[(d * HID + v + 8 * hl) * NFEAT + d];
        wpr[f] = Wp[d];
    }
    const float bpv = bp[0];

    const int tile0 = blockIdx.x * TPW;
    #pragma unroll 1
    for (int t = 0; t < TPW; ++t) {
        const int s0 = (tile0 + t) * 16;

        // ---- stage x tile (16 contiguous rows = 4KB) into bank-padded LDS ----
        {
            const int e0 = tid, e1 = tid + 512;
            xs[(e0 >> 6) * ROWPAD + (e0 & 63)] = x[s0 * NFEAT + e0];
            xs[(e1 >> 6) * ROWPAD + (e1 & 63)] = x[s0 * NFEAT + e1];
            if (tid < 16) preds[tid] = 0.0f;
            if (t + 1 < TPW)  // hint next tile into cache (global_prefetch_b8)
                __builtin_prefetch(&x[(s0 + 16) * NFEAT + tid], 0, 1);
        }
        __syncthreads();

        float pacc = 0.0f;
        #pragma unroll
        for (int f = 0; f < FPW; ++f) {
            const int d = wv * FPW + f;
            const float xv = xs[ln * ROWPAD + d];   // x[sample=ln, feature d] broadcast

            // layer 1: o1^T = elu(w1 outer x), built directly in B-fragment layout
            v2f b[4];
            #pragma unroll
            for (int k = 0; k < 4; ++k) {
                b[k][0] = eluf(w1r[f][2 * k + 0] * xv);
                b[k][1] = eluf(w1r[f][2 * k + 1] * xv);
            }

            // layer 2: o2^T = W2^T * o1^T  (4x chained f32 16x16x4 WMMA, K=16)
            v8f c = {0.f, 0.f, 0.f, 0.f, 0.f, 0.f, 0.f, 0.f};
            #pragma unroll
            for (int k = 0; k < 4; ++k)
                c = __builtin_amdgcn_wmma_f32_16x16x4_f32(
                        false, w2a[f][k], false, b[k], (short)0, c, false, false);

            // ELU, then C-layout -> B-fragment layout via half-wave swaps
            float ce[8], sw[8];
            #pragma unroll
            for (int v = 0; v < 8; ++v) { ce[v] = eluf(c[v]); }
            #pragma unroll
            for (int v = 0; v < 8; ++v) { sw[v] = swap16(ce[v]); }
            const bool lo = (hl == 0);
            b[0][0] = lo ? ce[0] : sw[2];   // K=0..3 frag
            b[0][1] = lo ? ce[1] : sw[3];
            b[1][0] = lo ? ce[4] : sw[6];   // K=4..7 frag
            b[1][1] = lo ? ce[5] : sw[7];
            b[2][0] = lo ? sw[0] : ce[2];   // K=8..11 frag
            b[2][1] = lo ? sw[1] : ce[3];
            b[3][0] = lo ? sw[4] : ce[6];   // K=12..15 frag
            b[3][1] = lo ? sw[5] : ce[7];

            // layer 3: o3^T = W3^T * o2^T
            v8f c3 = {0.f, 0.f, 0.f, 0.f, 0.f, 0.f, 0.f, 0.f};
            #pragma unroll
            for (int k = 0; k < 4; ++k)
                c3 = __builtin_amdgcn_wmma_f32_16x16x4_f32(
                         false, w3a[f][k], false, b[k], (short)0, c3, false, false);

            // layer 4: out[d, s] = sum_h w4[h] * elu(o3^T[h, s])  (8 FMA + swap-reduce)
            float acc = 0.0f;
            #pragma unroll
            for (int v = 0; v < 8; ++v)
                acc = fmaf(w4r[f][v], eluf(c3[v]), acc);
            acc += swap16(acc);             // all lanes now hold out[d, s=ln]

            if (lane < 16) os[ln * ROWPAD + d] = acc;   // banks (4*ln+d)%64: conflict-free
            pacc = fmaf(acc, wpr[f], pacc);             // pred partial for this wave
        }
        if (lane < 16) atomicAdd(&preds[ln], pacc);     // ds_add_f32
        __syncthreads();

        // ---- coalesced writeback of out tile (+ pred) ----
        {
            const int e0 = tid, e1 = tid + 512;
            gam_out[s0 * NFEAT + e0] = os[(e0 >> 6) * ROWPAD + (e0 & 63)];
            gam_out[s0 * NFEAT + e1] = os[(e1 >> 6) * ROWPAD + (e1 & 63)];
            if (tid < 16) pred_out[s0 + tid] = preds[tid] + bpv;
        }
        __syncthreads();
    }
}

extern "C" void kernel_launch(void* const* d_in, const int* in_sizes, int n_in,
                              void* d_out, int out_size, void* d_ws, size_t ws_size,
                              hipStream_t stream) {
    (void)in_sizes; (void)n_in; (void)d_ws; (void)ws_size; (void)out_size;
    const float* x  = (const float*)d_in[0];
    const float* W1 = (const float*)d_in[1];
    const float* W2 = (const float*)d_in[2];
    const float* W3 = (const float*)d_in[3];
    const float* W4 = (const float*)d_in[4];
    const float* Wp = (const float*)d_in[5];
    const float* bp = (const float*)d_in[6];
    float* pred_out = (float*)d_out;            // (N,1) first in tuple order
    float* gam_out  = pred_out + NSAMP;         // (N,64) second
    gam_wmma_kernel<<<dim3(NBLK), dim3(512), 0, stream>>>(
        x, W1, W2, W3, W4, Wp, bp, pred_out, gam_out);
}